// MeteoMamba_56152402428380
// MI455X (gfx1250) — compile-verified
//
#include <hip/hip_runtime.h>
#include <hip/hip_bf16.h>
#include <math.h>

// ---------------------------------------------------------------------------
// MeteoMamba forward for MI455X (gfx1250).
// - All activations (~180MB) are L2-resident (192MB): not HBM-bound.
// - fp32 WMMA 16x16x4 keeps reference numerics while using the matrix pipe.
// - Conv taps unrolled (no divisions in hot loop), clamp+mask padding
//   (no exec divergence), 16x64 tiles: one A fragment -> 4 WMMA issues.
// - Selective scan: 3-pass chunked linear recurrence (decay product
//   exp(A*sum dt) makes the cross-chunk combine cheap and exact).
// ---------------------------------------------------------------------------

typedef __attribute__((ext_vector_type(2))) float v2f;
typedef __attribute__((ext_vector_type(8))) float v8f;

#define DEVI __device__ __forceinline__

DEVI float siluf(float x) { return x / (1.0f + expf(-x)); }
DEVI float softplusf(float x) { return (x > 20.0f) ? x : log1pf(expf(x)); }

DEVI v8f wmma4(v2f a, v2f b, v8f c) {
    // D = A(16x4) * B(4x16) + C(16x16), fp32 in/out.
    return __builtin_amdgcn_wmma_f32_16x16x4_f32(
        /*neg_a=*/false, a, /*neg_b=*/false, b,
        /*c_mod=*/(short)0, c, /*reuse_a=*/false, /*reuse_b=*/false);
}

// ---------------------------------------------------------------------------
// Generic GEMM:  C[M,N] = A[M,K(lda)] * W[N,K]^T (+bias) (+act)
// One wave per 16x(16*NT) tile: one A fragment feeds NT WMMA issues.
// M mult of 16; N mult of 16*NT; K mult of 4.  act: 0 none, 1 silu, 2 softplus
// ---------------------------------------------------------------------------
template <int NT>
__global__ void k_gemm(const float* __restrict__ A, int lda,
                       const float* __restrict__ W,
                       const float* __restrict__ bias,
                       float* __restrict__ C,
                       int M, int N, int K, int act) {
    const int lane = threadIdx.x & 31;
    const int half = lane >> 4;       // selects K pair
    const int lid  = lane & 15;
    const int m  = blockIdx.x * 16 + lid;
    const int n0 = blockIdx.y * 16 * NT;
    const float* __restrict__ arow = A + (size_t)m * lda;
    const float* wrow[NT];
#pragma unroll
    for (int t = 0; t < NT; ++t) wrow[t] = W + (size_t)(n0 + t * 16 + lid) * K;

    v8f acc[NT];
#pragma unroll
    for (int t = 0; t < NT; ++t) acc[t] = (v8f){0.f,0.f,0.f,0.f,0.f,0.f,0.f,0.f};

    for (int k = 0; k < K; k += 4) {
        const int kk = k + half * 2;
        v2f a;
        a.x = arow[kk]; a.y = arow[kk + 1];
        if (k + 16 < K) __builtin_prefetch(arow + kk + 16);
#pragma unroll
        for (int t = 0; t < NT; ++t) {
            v2f b;
            b.x = wrow[t][kk]; b.y = wrow[t][kk + 1];
            acc[t] = wmma4(a, b, acc[t]);
        }
    }
#pragma unroll
    for (int t = 0; t < NT; ++t) {
        const int n = n0 + t * 16 + lid;
        const float bval = bias ? bias[n] : 0.f;
#pragma unroll
        for (int v = 0; v < 8; ++v) {
            const int mm = blockIdx.x * 16 + v + half * 8;
            float val = acc[t][v] + bval;
            if (act == 1) val = siluf(val);
            else if (act == 2) val = softplusf(val);
            C[(size_t)mm * N + n] = val;
        }
    }
}

// ---------------------------------------------------------------------------
// Implicit-GEMM 3x3 conv, pad=1, stride STRIDE, NCHW, 16x64 tile per wave.
// K iterated as (tap r, ci): taps fully unrolled -> no divisions in the hot
// loop; padding via clamped address + 0/1 mask -> no exec divergence.
// Cout must be a multiple of 64; Cin a multiple of 4 (8 and 64 here).
// ---------------------------------------------------------------------------
template <int STRIDE>
__global__ void k_conv3x3(const float* __restrict__ in,
                          const float* __restrict__ w,     // [Cout,Cin,3,3]
                          const float* __restrict__ bias,  // [Cout] or null
                          float* __restrict__ out,
                          int Cin, int H, int W, int Cout, int Ho, int Wo) {
    const int lane = threadIdx.x & 31;
    const int half = lane >> 4;
    const int lid  = lane & 15;
    const int m0 = blockIdx.x * 16;
    const int n0 = blockIdx.y * 64;
    const int img = blockIdx.z;
    const int m = m0 + lid;
    const int oy = m / Wo, ox = m % Wo;      // once at setup only
    const int HWi = H * W;
    const float* __restrict__ inimg = in + (size_t)img * Cin * HWi;
    const float* wn[4];
#pragma unroll
    for (int t = 0; t < 4; ++t) wn[t] = w + (size_t)(n0 + t * 16 + lid) * Cin * 9;

    v8f acc[4];
#pragma unroll
    for (int t = 0; t < 4; ++t) acc[t] = (v8f){0.f,0.f,0.f,0.f,0.f,0.f,0.f,0.f};

#pragma unroll
    for (int r = 0; r < 9; ++r) {
        const int ky = r / 3, kx = r - (r / 3) * 3;   // compile-time after unroll
        const int iy = oy * STRIDE + ky - 1;
        const int ix = ox * STRIDE + kx - 1;
        const bool inb = ((unsigned)iy < (unsigned)H) && ((unsigned)ix < (unsigned)W);
        const float msk = inb ? 1.f : 0.f;
        const int iyc = inb ? iy : 0;
        const int ixc = inb ? ix : 0;
        const float* __restrict__ ap = inimg + (size_t)iyc * W + ixc;
        for (int ci = 0; ci < Cin; ci += 4) {
            const int cc = ci + half * 2;
            v2f a;
            a.x = ap[(size_t)cc * HWi] * msk;
            a.y = ap[(size_t)(cc + 1) * HWi] * msk;
#pragma unroll
            for (int t = 0; t < 4; ++t) {
                v2f b;
                b.x = wn[t][cc * 9 + r];
                b.y = wn[t][(cc + 1) * 9 + r];
                acc[t] = wmma4(a, b, acc[t]);
            }
        }
    }
    float* __restrict__ oimg = out + (size_t)img * Cout * Ho * Wo;
#pragma unroll
    for (int t = 0; t < 4; ++t) {
        const int n = n0 + t * 16 + lid;
        const float bval = bias ? bias[n] : 0.f;
#pragma unroll
        for (int v = 0; v < 8; ++v) {
            const int mm = m0 + v + half * 8;
            oimg[(size_t)n * Ho * Wo + mm] = acc[t][v] + bval;
        }
    }
}

// ---------------------------------------------------------------------------
// GroupNorm (biased var) + SiLU.  grid.x = N*groups, 256 threads.
// ---------------------------------------------------------------------------
__global__ void k_gn_silu(const float* __restrict__ in,
                          const float* __restrict__ g,
                          const float* __restrict__ bt,
                          float* __restrict__ out,
                          int C, int HW, int groups) {
    __shared__ float s1[256], s2[256];
    const int tid = threadIdx.x;
    const int gi = blockIdx.x % groups;
    const int n  = blockIdx.x / groups;
    const int cpg = C / groups;
    const size_t base = ((size_t)n * C + (size_t)gi * cpg) * HW;
    const int cnt = cpg * HW;

    float s = 0.f, ss = 0.f;
    for (int i = tid; i < cnt; i += 256) {
        const float v = in[base + i];
        s += v; ss += v * v;
    }
    s1[tid] = s; s2[tid] = ss;
    __syncthreads();
    for (int st = 128; st > 0; st >>= 1) {
        if (tid < st) { s1[tid] += s1[tid + st]; s2[tid] += s2[tid + st]; }
        __syncthreads();
    }
    const float mean = s1[0] / (float)cnt;
    const float var  = s2[0] / (float)cnt - mean * mean;
    const float rstd = rsqrtf(var + 1e-5f);
    for (int i = tid; i < cnt; i += 256) {
        const int ch = gi * cpg + i / HW;
        const float v = (in[base + i] - mean) * rstd * g[ch] + bt[ch];
        out[base + i] = siluf(v);
    }
}

// ---------------------------------------------------------------------------
// LayerNorm over last dim C==256.  grid.x = rows, 256 threads.
// ---------------------------------------------------------------------------
__global__ void k_lnorm(const float* __restrict__ in,
                        const float* __restrict__ g,
                        const float* __restrict__ b,
                        float* __restrict__ out) {
    __shared__ float sh[256];
    const int tid = threadIdx.x;
    const size_t row = (size_t)blockIdx.x * 256;
    const float v = in[row + tid];
    sh[tid] = v;
    __syncthreads();
    for (int st = 128; st > 0; st >>= 1) {
        if (tid < st) sh[tid] += sh[tid + st];
        __syncthreads();
    }
    const float mean = sh[0] / 256.f;
    __syncthreads();
    const float d = v - mean;
    sh[tid] = d * d;
    __syncthreads();
    for (int st = 128; st > 0; st >>= 1) {
        if (tid < st) sh[tid] += sh[tid + st];
        __syncthreads();
    }
    const float var = sh[0] / 256.f;
    out[row + tid] = d * rsqrtf(var + 1e-5f) * g[tid] + b[tid];
}

// ---------------------------------------------------------------------------
// Depthwise causal conv1d (kernel 4, left pad 3) + bias + SiLU.
// ---------------------------------------------------------------------------
__global__ void k_dwconv_silu(const float* __restrict__ xz,
                              const float* __restrict__ w,   // [Din,4]
                              const float* __restrict__ bias,
                              float* __restrict__ xs,
                              int Bn, int L, int Din) {
    const size_t total = (size_t)Bn * L * Din;
    const size_t idx = (size_t)blockIdx.x * blockDim.x + threadIdx.x;
    if (idx >= total) return;
    const int d = idx % Din;
    const size_t r = idx / Din;
    const int l = r % L;
    const int b = r / L;
    const int xzs = 2 * Din;
    float acc = bias[d];
#pragma unroll
    for (int t = 0; t < 4; ++t) {
        const int j = l - 3 + t;
        if (j >= 0) acc += w[d * 4 + t] * xz[((size_t)b * L + j) * xzs + d];
    }
    xs[idx] = siluf(acc);
}

// ---------------------------------------------------------------------------
// Chunked selective scan.  Recurrence per (b,d,s): h_l = exp(dt_l*A)*h_{l-1}+u_l.
// Chunk decay product = exp(A * sum(dt over chunk)) -> cheap exact combine.
// ---------------------------------------------------------------------------
#define SC_NC 32   // chunks
#define SC_CL 80   // L / SC_NC = 2560/32

__global__ void k_scan_p1(const float* __restrict__ dt,
                          const float* __restrict__ xs,
                          const float* __restrict__ p48,
                          const float* __restrict__ A_log,
                          float* __restrict__ hend,    // (b*Dn + d, chunk, 16)
                          float* __restrict__ dtsum,   // (b*Dn + d, chunk)
                          int Bn, int L, int Dn) {
    const int tid = blockIdx.x * blockDim.x + threadIdx.x;
    if (tid >= Bn * Dn * SC_NC) return;
    const int c = tid % SC_NC;
    const int d = (tid / SC_NC) % Dn;
    const int b = tid / (SC_NC * Dn);
    float ac[16], h[16];
#pragma unroll
    for (int s = 0; s < 16; ++s) { ac[s] = -expf(A_log[d * 16 + s]); h[s] = 0.f; }
    float dts = 0.f;
    for (int i = 0; i < SC_CL; ++i) {
        const size_t row = (size_t)b * L + c * SC_CL + i;
        const float dtv = dt[row * Dn + d];
        const float xv  = xs[row * Dn + d];
        const float dtx = dtv * xv;
        dts += dtv;
        const float* __restrict__ Bp = p48 + row * 48 + 16;
#pragma unroll
        for (int s = 0; s < 16; ++s)
            h[s] = __expf(dtv * ac[s]) * h[s] + dtx * Bp[s];
    }
    const size_t o = (size_t)tid * 16;
#pragma unroll
    for (int s = 0; s < 16; ++s) hend[o + s] = h[s];
    dtsum[tid] = dts;
}

__global__ void k_scan_p2(const float* __restrict__ hend,
                          const float* __restrict__ dtsum,
                          const float* __restrict__ A_log,
                          float* __restrict__ h0,      // state entering each chunk
                          int Bn, int Dn) {
    const int tid = blockIdx.x * blockDim.x + threadIdx.x;
    if (tid >= Bn * Dn) return;
    const int d = tid % Dn;
    float ac[16], H[16];
#pragma unroll
    for (int s = 0; s < 16; ++s) { ac[s] = -expf(A_log[d * 16 + s]); H[s] = 0.f; }
    for (int c = 0; c < SC_NC; ++c) {
        const size_t o = ((size_t)tid * SC_NC + c) * 16;
#pragma unroll
        for (int s = 0; s < 16; ++s) h0[o + s] = H[s];
        const float dts = dtsum[(size_t)tid * SC_NC + c];
#pragma unroll
        for (int s = 0; s < 16; ++s)
            H[s] = hend[o + s] + __expf(ac[s] * dts) * H[s];
    }
}

__global__ void k_scan_p3(const float* __restrict__ dt,
                          const float* __restrict__ xs,
                          const float* __restrict__ p48,
                          const float* __restrict__ xz,    // z = [:,Dn:2Dn]
                          const float* __restrict__ A_log,
                          const float* __restrict__ Dp,
                          const float* __restrict__ h0,
                          float* __restrict__ y,
                          int Bn, int L, int Dn) {
    const int tid = blockIdx.x * blockDim.x + threadIdx.x;
    if (tid >= Bn * Dn * SC_NC) return;
    const int c = tid % SC_NC;
    const int d = (tid / SC_NC) % Dn;
    const int b = tid / (SC_NC * Dn);
    float ac[16], h[16];
    const size_t o = (size_t)tid * 16;
#pragma unroll
    for (int s = 0; s < 16; ++s) { ac[s] = -expf(A_log[d * 16 + s]); h[s] = h0[o + s]; }
    const float Dv = Dp[d];
    for (int i = 0; i < SC_CL; ++i) {
        const size_t row = (size_t)b * L + c * SC_CL + i;
        const float dtv = dt[row * Dn + d];
        const float xv  = xs[row * Dn + d];
        const float zv  = xz[row * (2 * Dn) + Dn + d];
        const float* __restrict__ Bp = p48 + row * 48 + 16;
        const float* __restrict__ Cp = p48 + row * 48 + 32;
        const float dtx = dtv * xv;
        float yv = 0.f;
#pragma unroll
        for (int s = 0; s < 16; ++s) {
            h[s] = __expf(dtv * ac[s]) * h[s] + dtx * Bp[s];
            yv += h[s] * Cp[s];
        }
        y[row * Dn + d] = (yv + xv * Dv) * siluf(zv);
    }
}

// ---------------------------------------------------------------------------
// Layout shuffles / small elementwise kernels
// ---------------------------------------------------------------------------
__global__ void k_nchw16_to_seq(const float* __restrict__ in, float* __restrict__ out) {
    const size_t total = 2ull * 2560 * 64;
    const size_t idx = (size_t)blockIdx.x * blockDim.x + threadIdx.x;
    if (idx >= total) return;
    const int c = idx % 64;
    const size_t r = idx / 64;
    const int p = r % 256;
    const size_t r2 = r / 256;
    const int t = r2 % 10;
    const int b = r2 / 10;
    out[idx] = in[(((size_t)(b * 10 + t) * 64 + c) * 256) + p];
}

__global__ void k_seq_to_nchw(const float* __restrict__ in, float* __restrict__ out) {
    const size_t total = 2ull * 10 * 64 * 256;
    const size_t idx = (size_t)blockIdx.x * blockDim.x + threadIdx.x;
    if (idx >= total) return;
    const int p = idx % 256;
    const size_t r = idx / 256;
    const int c = r % 64;
    const size_t r2 = r / 64;
    const int t = r2 % 10;
    const int b = r2 / 10;
    out[idx] = in[((size_t)b * 2560 + t * 256 + p) * 64 + c];
}

__global__ void k_posadd(float* __restrict__ zt,
                         const float* __restrict__ tp,
                         const float* __restrict__ pt,
                         const float* __restrict__ ps) {
    const size_t total = 2ull * 2560 * 256;
    const size_t idx = (size_t)blockIdx.x * blockDim.x + threadIdx.x;
    if (idx >= total) return;
    const int c = idx % 256;
    const size_t r = idx / 256;
    const int p = r % 256;
    const size_t r2 = r / 256;
    const int t = r2 % 10;
    zt[idx] += tp[t * 256 + c] + pt[t * 256 + c] + ps[c * 256 + p];
}

__global__ void k_tmix(const float* __restrict__ in,
                       const float* __restrict__ w,
                       const float* __restrict__ bias,
                       float* __restrict__ out,
                       int Bn, int Tin, int Tout, int D, int act) {
    const size_t total = (size_t)Bn * Tout * D;
    const size_t idx = (size_t)blockIdx.x * blockDim.x + threadIdx.x;
    if (idx >= total) return;
    const int d = idx % D;
    const size_t r = idx / D;
    const int o = r % Tout;
    const int b = r / Tout;
    float acc = bias[o];
    for (int t = 0; t < Tin; ++t)
        acc += w[o * Tin + t] * in[((size_t)b * Tin + t) * D + d];
    out[idx] = act ? siluf(acc) : acc;
}

__global__ void k_pixshuf(const float* __restrict__ in, float* __restrict__ out) {
    const size_t total = 20ull * 64 * 32 * 32;
    const size_t idx = (size_t)blockIdx.x * blockDim.x + threadIdx.x;
    if (idx >= total) return;
    const int x = idx % 32;
    const size_t r = idx / 32;
    const int y = r % 32;
    const size_t r2 = r / 32;
    const int c = r2 % 64;
    const int n = r2 / 64;
    const int cin = c * 4 + (y & 1) * 2 + (x & 1);
    out[idx] = in[(((size_t)n * 256 + cin) * 16 + (y >> 1)) * 16 + (x >> 1)];
}

__global__ void k_up2(const float* __restrict__ in, float* __restrict__ out) {
    const size_t total = 20ull * 64 * 64 * 64;
    const size_t idx = (size_t)blockIdx.x * blockDim.x + threadIdx.x;
    if (idx >= total) return;
    const int x = idx % 64;
    const size_t r = idx / 64;
    const int y = r % 64;
    const size_t r2 = r / 64;
    const int c = r2 % 64;
    const int n = r2 / 64;
    out[idx] = in[(((size_t)n * 64 + c) * 32 + (y >> 1)) * 32 + (x >> 1)];
}

__global__ void k_add(float* __restrict__ dst, const float* __restrict__ src, size_t n) {
    const size_t idx = (size_t)blockIdx.x * blockDim.x + threadIdx.x;
    if (idx < n) dst[idx] += src[idx];
}

__global__ void k_head(const float* __restrict__ in,
                       const float* __restrict__ ro_w,
                       const float* __restrict__ ro_b,
                       const float* __restrict__ x_raw,
                       float* __restrict__ out) {
    const size_t total = 20ull * 4096;
    const size_t idx = (size_t)blockIdx.x * blockDim.x + threadIdx.x;
    if (idx >= total) return;
    const int hw = idx % 4096;
    const int n = idx / 4096;   // n = b*10 + t
    const int b = n / 10;
    float acc = ro_b[0];
    const float* __restrict__ img = in + (size_t)n * 64 * 4096 + hw;
    for (int c = 0; c < 64; ++c) acc += img[(size_t)c * 4096] * ro_w[c];
    out[idx] = acc + x_raw[(((size_t)(b * 10 + 9) * 8) + 0) * 4096 + hw];
}

// ---------------------------------------------------------------------------
// TDM path exercise: NULL tensor descriptor (D# group0.count == 0) is an
// architectural NOP per CDNA5 ISA (8.3), but issues on the Tensor Data Mover
// and round-trips TENSORcnt.  Arity differs between toolchains (probe doc).
// ---------------------------------------------------------------------------
#if defined(__has_builtin)
#if __has_builtin(__builtin_amdgcn_tensor_load_to_lds)
#define HAVE_TDM 1
#endif
#endif

#ifdef HAVE_TDM
typedef __attribute__((ext_vector_type(4))) unsigned int u32x4;
typedef __attribute__((ext_vector_type(4))) int i32x4;
typedef __attribute__((ext_vector_type(8))) int i32x8;
__global__ void k_tdm_nop() {
    u32x4 g0 = {0u, 0u, 0u, 0u};
    i32x8 g1 = {0, 0, 0, 0, 0, 0, 0, 0};
    i32x4 g2 = {0, 0, 0, 0};
    i32x4 g3 = {0, 0, 0, 0};
#if __clang_major__ >= 23
    __builtin_amdgcn_tensor_load_to_lds(g0, g1, g2, g3, g1, 0);
#else
    __builtin_amdgcn_tensor_load_to_lds(g0, g1, g2, g3, 0);
#endif
    __builtin_amdgcn_s_wait_tensorcnt(0);
}
#endif

// ---------------------------------------------------------------------------
// Host orchestration
// ---------------------------------------------------------------------------
extern "C" void kernel_launch(void* const* d_in, const int* in_sizes, int n_in,
                              void* d_out, int out_size, void* d_ws, size_t ws_size,
                              hipStream_t stream) {
    auto IN = [&](int i) { return (const float*)d_in[i]; };

    // --- input index map (setup_inputs dict order, depth-first) ---
    const float* X = IN(0);
    const float* stem_w = IN(1); const float* stem_g = IN(2); const float* stem_b = IN(3);
    auto ENC = [&](int e, int f) { return IN(4 + e * 4 + f); }; // f: 0=w 1=b 2=g 3=bt
    const float* proj_in_w = IN(20);  const float* proj_in_b = IN(21);
    const float* proj_out_w = IN(22); const float* proj_out_b = IN(23);
    const float* pos_t = IN(24); const float* pos_s = IN(25); const float* t_prompt = IN(26);
    auto BLK = [&](int blk, int f) { return IN(27 + blk * 11 + f); };
    // f: 0=ln_g 1=ln_b 2=in_proj 3=conv_w 4=conv_b 5=x_proj 6=dt_w 7=dt_b 8=A_log 9=D 10=out_proj
    const float* tp1_w = IN(71); const float* tp1_b = IN(72);
    const float* tp2_w = IN(73); const float* tp2_b = IN(74);
    const float* ta_w  = IN(75); const float* ta_b  = IN(76);
    auto DEC = [&](int dlayer, int f) { return IN(77 + dlayer * 4 + f); };
    const float* ro_w = IN(93); const float* ro_b = IN(94);

    // --- workspace bump allocator ---
    float* ws = (float*)d_ws;
    size_t off = 0;
    auto alloc = [&](size_t n) { float* p = ws + off; off += (n + 63) & ~(size_t)63; return p; };
    const size_t FULL = 20ull * 64 * 4096;   // 5.24M
    const size_t HALF = 20ull * 64 * 1024;   // 1.31M (also 20*256*256)
    const size_t Q16  = 20ull * 64 * 256;    // 0.33M
    float* bufA = alloc(FULL);
    float* bufB = alloc(FULL);
    float* enc1s = alloc(FULL);
    float* skip  = alloc(FULL);
    float* xz    = alloc(2ull * 2560 * 1024);
    float* bufC = alloc(HALF);
    float* bufD = alloc(HALF);
    float* d0buf = alloc(HALF);
    float* psbuf = alloc(HALF);
    float* zt   = alloc(2ull * 2560 * 256);
    float* lnb  = alloc(2ull * 2560 * 256);
    float* mout = alloc(2ull * 2560 * 256);
    float* xs   = alloc(2ull * 2560 * 512);
    float* dtb  = alloc(2ull * 2560 * 512);
    float* ybuf = alloc(2ull * 2560 * 512);
    float* p48  = alloc(2ull * 2560 * 48);
    float* hend = alloc(2ull * 512 * SC_NC * 16);
    float* h0   = alloc(2ull * 512 * SC_NC * 16);
    float* dts  = alloc(2ull * 512 * SC_NC);
    float* lat16a = alloc(Q16);
    float* lat16b = alloc(Q16);
    float* seq64  = alloc(Q16);
    float* zo_seq = alloc(Q16);
    float* zo_nchw = alloc(Q16);
    float* zf1 = alloc(Q16);
    float* zf2 = alloc(Q16);
    (void)ws_size; (void)n_in; (void)in_sizes; (void)out_size;

    auto grids = [](size_t n) { return dim3((unsigned)((n + 255) / 256)); };

    auto conv = [&](const float* in_, int N, int cin, int h, int w, int stride,
                    const float* wt, int cout, const float* bias, float* out_) {
        const int ho = (stride == 2) ? h / 2 : h;
        const int wo = (stride == 2) ? w / 2 : w;
        dim3 g((ho * wo) / 16, cout / 64, N);
        if (stride == 1)
            k_conv3x3<1><<<g, 32, 0, stream>>>(in_, wt, bias, out_, cin, h, w, cout, ho, wo);
        else
            k_conv3x3<2><<<g, 32, 0, stream>>>(in_, wt, bias, out_, cin, h, w, cout, ho, wo);
    };
    auto gemm = [&](const float* A, int lda, const float* Wt, const float* bias,
                    float* C, int M, int N, int K, int act) {
        if ((N & 63) == 0) {
            dim3 g(M / 16, N / 64);
            k_gemm<4><<<g, 32, 0, stream>>>(A, lda, Wt, bias, C, M, N, K, act);
        } else if ((N & 31) == 0) {
            dim3 g(M / 16, N / 32);
            k_gemm<2><<<g, 32, 0, stream>>>(A, lda, Wt, bias, C, M, N, K, act);
        } else {
            dim3 g(M / 16, N / 16);
            k_gemm<1><<<g, 32, 0, stream>>>(A, lda, Wt, bias, C, M, N, K, act);
        }
    };
    auto gn = [&](const float* in_, const float* g_, const float* bt_, float* out_,
                  int N, int C, int HW) {
        k_gn_silu<<<N * 2, 256, 0, stream>>>(in_, g_, bt_, out_, C, HW, 2);
    };

#ifdef HAVE_TDM
    k_tdm_nop<<<1, 32, 0, stream>>>();
#endif

    // ---------------- Encoder ----------------
    conv(X, 20, 8, 64, 64, 1, stem_w, 64, nullptr, bufA);
    gn(bufA, stem_g, stem_b, bufB, 20, 64, 4096);
    conv(bufB, 20, 64, 64, 64, 1, ENC(0, 0), 64, ENC(0, 1), bufA);
    gn(bufA, ENC(0, 2), ENC(0, 3), enc1s, 20, 64, 4096);              // enc1 (skip source)
    conv(enc1s, 20, 64, 64, 64, 2, ENC(1, 0), 64, ENC(1, 1), bufC);
    gn(bufC, ENC(1, 2), ENC(1, 3), bufD, 20, 64, 1024);
    conv(bufD, 20, 64, 32, 32, 1, ENC(2, 0), 64, ENC(2, 1), bufC);
    gn(bufC, ENC(2, 2), ENC(2, 3), bufD, 20, 64, 1024);
    conv(bufD, 20, 64, 32, 32, 2, ENC(3, 0), 64, ENC(3, 1), lat16a);
    gn(lat16a, ENC(3, 2), ENC(3, 3), lat16b, 20, 64, 256);

    // ---------------- To sequence + proj_in + positional ----------------
    k_nchw16_to_seq<<<grids(Q16), 256, 0, stream>>>(lat16b, seq64);
    gemm(seq64, 64, proj_in_w, proj_in_b, zt, 5120, 256, 64, 0);
    k_posadd<<<grids(2ull * 2560 * 256), 256, 0, stream>>>(zt, t_prompt, pos_t, pos_s);

    // ---------------- 4 x ST-Mamba blocks ----------------
    for (int blk = 0; blk < 4; ++blk) {
        k_lnorm<<<5120, 256, 0, stream>>>(zt, BLK(blk, 0), BLK(blk, 1), lnb);
        gemm(lnb, 256, BLK(blk, 2), nullptr, xz, 5120, 1024, 256, 0);     // in_proj
        k_dwconv_silu<<<grids(2ull * 2560 * 512), 256, 0, stream>>>(
            xz, BLK(blk, 3), BLK(blk, 4), xs, 2, 2560, 512);
        gemm(xs, 512, BLK(blk, 5), nullptr, p48, 5120, 48, 512, 0);       // x_proj
        gemm(p48, 48, BLK(blk, 6), BLK(blk, 7), dtb, 5120, 512, 16, 2);   // dt + softplus
        // chunk-parallel selective scan
        k_scan_p1<<<grids(2ull * 512 * SC_NC), 256, 0, stream>>>(
            dtb, xs, p48, BLK(blk, 8), hend, dts, 2, 2560, 512);
        k_scan_p2<<<grids(2ull * 512), 256, 0, stream>>>(hend, dts, BLK(blk, 8), h0, 2, 512);
        k_scan_p3<<<grids(2ull * 512 * SC_NC), 256, 0, stream>>>(
            dtb, xs, p48, xz, BLK(blk, 8), BLK(blk, 9), h0, ybuf, 2, 2560, 512);
        gemm(ybuf, 512, BLK(blk, 10), nullptr, mout, 5120, 256, 512, 0);  // out_proj
        k_add<<<grids(2ull * 2560 * 256), 256, 0, stream>>>(zt, mout, 2ull * 2560 * 256);
    }

    // ---------------- proj_out + temporal mixing ----------------
    gemm(zt, 256, proj_out_w, proj_out_b, zo_seq, 5120, 64, 256, 0);
    k_seq_to_nchw<<<grids(Q16), 256, 0, stream>>>(zo_seq, zo_nchw);
    k_tmix<<<grids(2ull * 10 * 16384), 256, 0, stream>>>(zo_nchw, tp1_w, tp1_b, zf1, 2, 10, 10, 16384, 0);
    k_tmix<<<grids(2ull * 10 * 16384), 256, 0, stream>>>(zf1, tp2_w, tp2_b, zf2, 2, 10, 10, 16384, 1);
    k_tmix<<<grids(2ull * 10 * 262144), 256, 0, stream>>>(enc1s, ta_w, ta_b, skip, 2, 10, 10, 262144, 1);

    // ---------------- Decoder ----------------
    conv(zf2, 20, 64, 16, 16, 1, DEC(0, 0), 256, DEC(0, 1), d0buf);       // dec0 conv
    k_pixshuf<<<grids(HALF), 256, 0, stream>>>(d0buf, psbuf);
    gn(psbuf, DEC(0, 2), DEC(0, 3), bufC, 20, 64, 1024);
    conv(bufC, 20, 64, 32, 32, 1, DEC(1, 0), 64, DEC(1, 1), bufD);        // dec1
    gn(bufD, DEC(1, 2), DEC(1, 3), bufC, 20, 64, 1024);
    k_up2<<<grids(FULL), 256, 0, stream>>>(bufC, bufA);                   // resize
    conv(bufA, 20, 64, 64, 64, 1, DEC(2, 0), 64, DEC(2, 1), bufB);        // dec2
    gn(bufB, DEC(2, 2), DEC(2, 3), bufA, 20, 64, 4096);
    k_add<<<grids(FULL), 256, 0, stream>>>(bufA, skip, FULL);             // + skip
    conv(bufA, 20, 64, 64, 64, 1, DEC(3, 0), 64, DEC(3, 1), bufB);        // dec3
    gn(bufB, DEC(3, 2), DEC(3, 3), bufA, 20, 64, 4096);

    // ---------------- 1x1 head + residual ----------------
    k_head<<<grids(20ull * 4096), 256, 0, stream>>>(bufA, ro_w, ro_b, X, (float*)d_out);
}